// YoloLoss_34282428957023
// MI455X (gfx1250) — compile-verified
//
#include <hip/hip_runtime.h>
#include <hip/hip_bf16.h>
#include <stdint.h>

// ---------------- problem constants (from reference) ----------------
#define S7     7
#define PREDC  90            // B*5 + C = 2*5 + 80
#define TGTC   85            // 5 + C
#define NIMG   8192
#define CELLS  (NIMG * S7 * S7)   // 401408
#define TILE   64                 // cells per LDS tile (tile is contiguous in memory)
#define NTILES (CELLS / TILE)     // 6272 exactly
#define TPB    256
#define NWAVES (TPB / 32)

typedef __attribute__((ext_vector_type(4))) int v4i;
typedef __attribute__((address_space(1))) v4i* glob_v4i_ptr;
typedef __attribute__((address_space(3))) v4i* lds_v4i_ptr;
typedef __attribute__((address_space(3))) void* lds_void_ptr;

// ---- CDNA5 async global->LDS copy (ASYNCcnt path) ----
// cpol = 1 -> TH_NT: pure streaming input, read exactly once; don't retain in caches.
__device__ __forceinline__ void async_copy_b128(const float* g, float* l) {
#if __has_builtin(__builtin_amdgcn_global_load_async_to_lds_b128)
  __builtin_amdgcn_global_load_async_to_lds_b128((glob_v4i_ptr)g, (lds_v4i_ptr)l, 0, 1);
#else
  asm volatile("global_load_async_to_lds_b128 %0, %1, off th:TH_LOAD_NT"
               :
               : "v"((unsigned)(uintptr_t)(lds_void_ptr)l), "v"(g)
               : "memory");
#endif
}

__device__ __forceinline__ void wait_async_zero() {
#if __has_builtin(__builtin_amdgcn_s_wait_asynccnt)
  __builtin_amdgcn_s_wait_asynccnt(0);
#else
  asm volatile("s_wait_asynccnt 0" ::: "memory");
#endif
}

// IoU of (already-absolute) target box vs a relative pred box in cell (iy,jx)
__device__ __forceinline__ float box_iou(float tax1, float tay1, float tax2, float tay2,
                                         float x, float y, float w, float h,
                                         float jx, float iy) {
  float xc  = (x + jx) / 7.0f;
  float yc  = (y + iy) / 7.0f;
  float ax1 = xc - w * 0.5f;
  float ay1 = yc - h * 0.5f;
  float ax2 = xc + w * 0.5f;
  float ay2 = yc + h * 0.5f;
  float iw  = fmaxf(fminf(tax2, ax2) - fmaxf(tax1, ax1), 0.0f);
  float ih  = fmaxf(fminf(tay2, ay2) - fmaxf(tay1, ay1), 0.0f);
  float inter  = iw * ih;
  float area_t = (tax2 - tax1) * (tay2 - tay1);
  float area_p = (ax2 - ax1) * (ay2 - ay1);
  return inter / (area_t + area_p - inter + 1e-6f);
}

// Stage 1: per-block partial sums of the 5 loss components.
__global__ __launch_bounds__(TPB) void yolo_partial_kernel(const float* __restrict__ pred,
                                                           const float* __restrict__ tgt,
                                                           float* __restrict__ partial,
                                                           int ntiles) {
  __shared__ __align__(16) float lp[TILE * PREDC];   // 23040 B
  __shared__ __align__(16) float lt[TILE * TGTC];    // 21760 B
  __shared__ float red[NWAVES][5];

  float a_xy = 0.f, a_wh = 0.f, a_co = 0.f, a_cn = 0.f, a_cl = 0.f;
  const int tid = threadIdx.x;
  const int c   = tid >> 2;   // cell within tile: 0..63
  const int sub = tid & 3;    // 4 threads collaborate per cell

  for (int tile = blockIdx.x; tile < ntiles; tile += gridDim.x) {
    const long long cell0 = (long long)tile * TILE;
    const float* gp = pred + cell0 * PREDC;   // contiguous 5760-float chunk, 16B aligned
    const float* gt = tgt  + cell0 * TGTC;    // contiguous 5440-float chunk, 16B aligned

    for (int i = tid; i < TILE * PREDC / 4; i += TPB)   // 1440 x b128
      async_copy_b128(gp + i * 4, &lp[i * 4]);
    for (int i = tid; i < TILE * TGTC / 4; i += TPB)    // 1360 x b128
      async_copy_b128(gt + i * 4, &lt[i * 4]);
    wait_async_zero();
    __syncthreads();

    const float* P = &lp[c * PREDC];
    const float* T = &lt[c * TGTC];
    const float true_c = T[4];
    const float objf   = (true_c != 0.0f) ? 1.0f : 0.0f;

    // class SSE: this sub-thread handles 20 of the 80 channels
    float cls = 0.0f;
    const int kb = 20 * sub;
#pragma unroll
    for (int k = 0; k < 20; ++k) {
      float d = T[5 + kb + k] - P[10 + kb + k];
      cls = fmaf(d, d, cls);
    }
    a_cl += objf * cls;

    if (sub == 0) {  // box / conf terms, once per cell
      const long long gc = cell0 + c;
      const float jx = (float)(int)(gc % S7);
      const float iy = (float)(int)((gc / S7) % S7);

      const float tx = T[0], ty = T[1], tw = T[2], th = T[3];
      const float txc  = (tx + jx) / 7.0f;
      const float tyc  = (ty + iy) / 7.0f;
      const float tax1 = txc - tw * 0.5f, tay1 = tyc - th * 0.5f;
      const float tax2 = txc + tw * 0.5f, tay2 = tyc + th * 0.5f;

      const float iou0 = box_iou(tax1, tay1, tax2, tay2, P[0], P[1], P[2], P[3], jx, iy);
      const float iou1 = box_iou(tax1, tay1, tax2, tay2, P[5], P[6], P[7], P[8], jx, iy);
      const bool  sel  = iou0 > iou1;   // reference: where(iou0>iou1, box1, box0)
      const float bx = sel ? P[5] : P[0];
      const float by = sel ? P[6] : P[1];
      const float bw = sel ? P[7] : P[2];
      const float bh = sel ? P[8] : P[3];

      const float dx = tx - bx, dy = ty - by;

      const bool  obj  = (true_c != 0.0f);
      const float tw_s = obj ? tw : 1.0f;
      const float th_s = obj ? th : 1.0f;
      const float bw_s = obj ? fmaxf(bw, 0.0f) : 1.0f;
      const float bh_s = obj ? fmaxf(bh, 0.0f) : 1.0f;
      const float dw   = sqrtf(tw_s) - sqrtf(bw_s);
      const float dh   = sqrtf(th_s) - sqrtf(bh_s);

      const float pc  = P[9];                 // channel 4 + 5*(B-1)
      const float cd  = true_c - pc;
      const float cse = cd * cd;

      a_xy += objf * (dx * dx + dy * dy);
      a_wh += objf * (dw * dw + dh * dh);
      a_co += objf * cse;
      a_cn += (1.0f - objf) * cse;
    }
    __syncthreads();   // before next tile overwrites LDS
  }

  // wave32 tree reduction, then cross-wave via LDS (deterministic order)
#pragma unroll
  for (int m = 16; m >= 1; m >>= 1) {
    a_xy += __shfl_xor(a_xy, m, 32);
    a_wh += __shfl_xor(a_wh, m, 32);
    a_co += __shfl_xor(a_co, m, 32);
    a_cn += __shfl_xor(a_cn, m, 32);
    a_cl += __shfl_xor(a_cl, m, 32);
  }
  const int wave = tid >> 5;
  if ((tid & 31) == 0) {
    red[wave][0] = a_xy; red[wave][1] = a_wh; red[wave][2] = a_co;
    red[wave][3] = a_cn; red[wave][4] = a_cl;
  }
  __syncthreads();
  if (tid == 0) {
    float s0 = 0, s1 = 0, s2 = 0, s3 = 0, s4 = 0;
#pragma unroll
    for (int w = 0; w < NWAVES; ++w) {
      s0 += red[w][0]; s1 += red[w][1]; s2 += red[w][2];
      s3 += red[w][3]; s4 += red[w][4];
    }
    float* p = partial + (size_t)blockIdx.x * 5;
    p[0] = s0; p[1] = s1; p[2] = s2; p[3] = s3; p[4] = s4;
  }
}

// Stage 2: single-block deterministic reduction + final scalar math.
__global__ __launch_bounds__(TPB) void yolo_final_kernel(const float* __restrict__ partial,
                                                         int nb, float* __restrict__ out) {
  __shared__ float red[NWAVES][5];
  float s[5] = {0, 0, 0, 0, 0};
  for (int r = threadIdx.x; r < nb; r += TPB) {
    const float* p = partial + (size_t)r * 5;
#pragma unroll
    for (int q = 0; q < 5; ++q) s[q] += p[q];
  }
#pragma unroll
  for (int m = 16; m >= 1; m >>= 1) {
#pragma unroll
    for (int q = 0; q < 5; ++q) s[q] += __shfl_xor(s[q], m, 32);
  }
  const int wave = threadIdx.x >> 5;
  if ((threadIdx.x & 31) == 0) {
#pragma unroll
    for (int q = 0; q < 5; ++q) red[wave][q] = s[q];
  }
  __syncthreads();
  if (threadIdx.x == 0) {
    float t[5] = {0, 0, 0, 0, 0};
#pragma unroll
    for (int w = 0; w < NWAVES; ++w) {
#pragma unroll
      for (int q = 0; q < 5; ++q) t[q] += red[w][q];
    }
    const float bs  = (float)NIMG;
    const float lxy = t[0] / bs;
    const float lwh = t[1] / bs;
    const float lco = t[2] / bs;
    const float lcn = t[3] / bs;
    const float lcl = t[4] / bs;
    const float loss = (5.0f * lxy + 5.0f * lwh + lco + 0.5f * lcn + lcl) / bs;
    out[0] = loss; out[1] = lxy; out[2] = lwh;
    out[3] = lco;  out[4] = lcn; out[5] = lcl;
  }
}

extern "C" void kernel_launch(void* const* d_in, const int* in_sizes, int n_in,
                              void* d_out, int out_size, void* d_ws, size_t ws_size,
                              hipStream_t stream) {
  (void)in_sizes; (void)n_in; (void)out_size;
  const float* pred = (const float*)d_in[0];
  const float* tgt  = (const float*)d_in[1];
  float* out        = (float*)d_out;
  float* partial    = (float*)d_ws;

  int nb = 3136;  // 6272 tiles / 2 tiles per block
  if (ws_size < (size_t)nb * 5 * sizeof(float)) {
    nb = (int)(ws_size / (5 * sizeof(float)));
    if (nb < 1) nb = 1;
  }
  yolo_partial_kernel<<<nb, TPB, 0, stream>>>(pred, tgt, partial, NTILES);
  yolo_final_kernel<<<1, TPB, 0, stream>>>(partial, nb, out);
}